// LUTWeight_80032420594224
// MI455X (gfx1250) — compile-verified
//
#include <hip/hip_runtime.h>

// Pure LUT gather: out[p, :, :] = weight[i0[p], i1[p], i2[p], i3[p], :, :]
// weight: [17,17,17,17,4,4] f32 (5.35 MB, L2-resident), out: [N,4,4] f32 (256 MB stream).
// Memory-bound; roofline ~14 us at 23.3 TB/s. No FLOPs -> WMMA not applicable;
// optimize the CDNA5 memory path instead (b128 vmem, NT temporal hints, coalesced stores).

typedef float v4f __attribute__((ext_vector_type(4)));

#define L_DIM 17
#define L2 (L_DIM * L_DIM)        // 289
#define L3 (L_DIM * L_DIM * L_DIM) // 4913

__global__ __launch_bounds__(256)
void lut_gather_kernel(const int* __restrict__ i0,
                       const int* __restrict__ i1,
                       const int* __restrict__ i2,
                       const int* __restrict__ i3,
                       const float* __restrict__ weight,
                       float* __restrict__ out,
                       unsigned n_pixels) {
    unsigned tid = blockIdx.x * blockDim.x + threadIdx.x;
    unsigned p = tid >> 2;   // pixel id: 4 lanes cooperate on one 16-float row
    unsigned s = tid & 3;    // which float4 of the row this lane moves
    if (p >= n_pixels) return;

    // Four adjacent lanes read the same dword -> single cacheline request.
    // Index streams are read exactly once: non-temporal so they don't evict the LUT.
    int a = __builtin_nontemporal_load(i0 + p);
    int b = __builtin_nontemporal_load(i1 + p);
    int c = __builtin_nontemporal_load(i2 + p);
    int d = __builtin_nontemporal_load(i3 + p);

    unsigned idx = (unsigned)a * L3 + (unsigned)b * L2 + (unsigned)c * L_DIM + (unsigned)d;

    // Gather 16 B of the 64 B LUT row. Regular (RT) temporal hint: the 5.35 MB
    // table stays resident in the 192 MB L2, so random gathers hit L2, not HBM.
    v4f v = *(const v4f*)(weight + idx * 16u + s * 4u);

    // Output is a 256 MB write-once stream: NT store (th:TH_STORE_NT) so it
    // write-rinses through L2 instead of displacing the LUT.
    // Lane layout makes each wave's b128 store a contiguous 512 B -> fully coalesced.
    __builtin_nontemporal_store(v, (v4f*)(out + (size_t)p * 16u + s * 4u));
}

extern "C" void kernel_launch(void* const* d_in, const int* in_sizes, int n_in,
                              void* d_out, int out_size, void* d_ws, size_t ws_size,
                              hipStream_t stream) {
    const int*   i0 = (const int*)d_in[0];
    const int*   i1 = (const int*)d_in[1];
    const int*   i2 = (const int*)d_in[2];
    const int*   i3 = (const int*)d_in[3];
    const float* w  = (const float*)d_in[4];
    float*       out = (float*)d_out;

    unsigned n_pixels = (unsigned)in_sizes[0];          // N = 4,194,304
    unsigned n_threads = n_pixels * 4u;                 // 4 lanes per pixel
    unsigned block = 256;                               // 8 waves (wave32)
    unsigned grid = (n_threads + block - 1) / block;

    lut_gather_kernel<<<grid, block, 0, stream>>>(i0, i1, i2, i3, w, out, n_pixels);
}